// GlobalAttentionLayer_22024592294542
// MI455X (gfx1250) — compile-verified
//
#include <hip/hip_runtime.h>
#include <math.h>

// ---------------------------------------------------------------------------
// Global attention pooling, MI455X (gfx1250).
//   Phase 1 (bandwidth-bound, ~22us @ 23.3TB/s): fused [N,256]x[256,16] GEMM
//     via V_WMMA_F32_16X16X4_F32 producing per-node (gate, o0, o1).
//     - 2 M-tiles per wave (shared B fragments), K-batched loads for latency
//       hiding, nontemporal A stream (states read exactly once, > L2 size).
//   Phase 2: global max, size-scan, per-graph softmax-weighted reduction
//     over the 6 MB (g,o0,o1) workspace (L2-resident).
// ---------------------------------------------------------------------------

typedef __attribute__((ext_vector_type(2))) float v2f;
typedef __attribute__((ext_vector_type(8))) float v8f;

#define D_FEAT 256
#define NCOLS 16
#define LDS_STRIDE 260           // dwords; banks {4n,4n+1}/{4n+2,4n+3}, conflict-free
#define WAVES_PER_BLOCK 8
#define ROWS_PER_WAVE 32         // two 16-row WMMA tiles per wave
#define KBATCH 8                 // K-steps fetched ahead of the WMMA burst

// ---------------- Phase 1: fused gate+output GEMM via f32 WMMA ----------------
__global__ __launch_bounds__(256) void ga_pass1(
    const float* __restrict__ states,
    const float* __restrict__ gate_w, const float* __restrict__ gate_b,
    const float* __restrict__ out_w,  const float* __restrict__ out_b,
    float* __restrict__ gbuf, float* __restrict__ obuf, int n_nodes)
{
    // Weight panel Wc[256][16] (col0=gate_w, col1=out_w[:,0], col2=out_w[:,1],
    // cols 3..15 = 0), stored transposed in LDS as ldsW[n*LDS_STRIDE + k].
    __shared__ float ldsW[NCOLS * LDS_STRIDE];
    {
        int k = threadIdx.x;                       // 256 threads <-> 256 K rows
        ldsW[0 * LDS_STRIDE + k] = gate_w[k];
        ldsW[1 * LDS_STRIDE + k] = out_w[2 * k + 0];
        ldsW[2 * LDS_STRIDE + k] = out_w[2 * k + 1];
        #pragma unroll
        for (int n = 3; n < NCOLS; ++n) ldsW[n * LDS_STRIDE + k] = 0.f;
    }
    __syncthreads();

    const int wave = threadIdx.x >> 5;
    const int lane = threadIdx.x & 31;
    const int m    = lane & 15;          // M (for A) / N (for B,C)
    const int half = lane >> 4;          // lane-half selects K pair
    const long long row0 =
        ((long long)blockIdx.x * WAVES_PER_BLOCK + wave) * ROWS_PER_WAVE;
    if (row0 >= n_nodes) return;         // uniform per wave; after barrier

    // Clamp A row indices into range: an OOB row only pollutes the matching
    // row of D, whose store is guarded below. No divergent loads.
    const long long nm1 = (long long)n_nodes - 1;
    long long rA0 = row0 + m;        if (rA0 > nm1) rA0 = nm1;
    long long rA1 = row0 + 16 + m;   if (rA1 > nm1) rA1 = nm1;
    const float* ap0 = states + rA0 * D_FEAT + 2 * half;   // +4*kt per K step
    const float* ap1 = states + rA1 * D_FEAT + 2 * half;
    const float* bp  = &ldsW[m * LDS_STRIDE + 2 * half];   // +4*kt per K step

    v8f c0 = {}, c1 = {};
    for (int kb = 0; kb < D_FEAT / 4; kb += KBATCH) {
        v2f a0[KBATCH], a1[KBATCH], b[KBATCH];
        #pragma unroll
        for (int i = 0; i < KBATCH; ++i) {
            const int kt = kb + i;
            // A 16x4 f32 fragment: K(2*half), K(2*half+1); streamed once -> NT
            a0[i] = __builtin_nontemporal_load(
                        reinterpret_cast<const v2f*>(ap0 + 4 * kt));
            a1[i] = __builtin_nontemporal_load(
                        reinterpret_cast<const v2f*>(ap1 + 4 * kt));
            // B 4x16 f32 fragment: same K pair, column N=m (ds_load_b64)
            b[i]  = *reinterpret_cast<const v2f*>(bp + 4 * kt);
        }
        #pragma unroll
        for (int i = 0; i < KBATCH; ++i) {
            c0 = __builtin_amdgcn_wmma_f32_16x16x4_f32(
                     false, a0[i], false, b[i], (short)0, c0, false, false);
            c1 = __builtin_amdgcn_wmma_f32_16x16x4_f32(
                     false, a1[i], false, b[i], (short)0, c1, false, false);
        }
    }

    // C/D layout: lane (half,m) holds column N=m, rows M = 8*half + r.
    const float gb = gate_b[0], ob0 = out_b[0], ob1 = out_b[1];
    if (m < 3) {
        #pragma unroll
        for (int t = 0; t < 2; ++t) {
            #pragma unroll
            for (int r = 0; r < 8; ++r) {
                long long rowD = row0 + t * 16 + half * 8 + r;
                if (rowD < (long long)n_nodes) {
                    float v = (t == 0) ? c0[r] : c1[r];
                    if (m == 0)      gbuf[rowD]         = v + gb;
                    else if (m == 1) obuf[rowD * 2 + 0] = v + ob0;
                    else             obuf[rowD * 2 + 1] = v + ob1;
                }
            }
        }
    }
}

// ---------------- Phase 2a: global max of gate logits ----------------
__global__ __launch_bounds__(256) void ga_max_partial(
    const float* __restrict__ g, float* __restrict__ pmax, int n)
{
    __shared__ float s[256];
    float m = -3.4e38f;
    for (int i = blockIdx.x * 256 + threadIdx.x; i < n; i += 256 * 1024)
        m = fmaxf(m, g[i]);
    s[threadIdx.x] = m;
    __syncthreads();
    for (int o = 128; o > 0; o >>= 1) {
        if (threadIdx.x < o) s[threadIdx.x] = fmaxf(s[threadIdx.x], s[threadIdx.x + o]);
        __syncthreads();
    }
    if (threadIdx.x == 0) pmax[blockIdx.x] = s[0];
}

__global__ __launch_bounds__(256) void ga_max_final(
    const float* __restrict__ pmax, float* __restrict__ gmax, int nparts)
{
    __shared__ float s[256];
    float m = -3.4e38f;
    for (int i = threadIdx.x; i < nparts; i += 256) m = fmaxf(m, pmax[i]);
    s[threadIdx.x] = m;
    __syncthreads();
    for (int o = 128; o > 0; o >>= 1) {
        if (threadIdx.x < o) s[threadIdx.x] = fmaxf(s[threadIdx.x], s[threadIdx.x + o]);
        __syncthreads();
    }
    if (threadIdx.x == 0) gmax[0] = s[0];
}

// ---------------- Phase 2b: exclusive scan of graph sizes ----------------
__global__ __launch_bounds__(256) void ga_scan(
    const int* __restrict__ sizes, int* __restrict__ offs, int n_graphs)
{
    __shared__ int s[256];
    int t = threadIdx.x;
    int per = (n_graphs + 255) / 256;
    int beg = t * per;
    int end = beg + per; if (end > n_graphs) end = n_graphs;
    int sum = 0;
    for (int i = beg; i < end; ++i) sum += sizes[i];
    s[t] = sum;
    __syncthreads();
    for (int off = 1; off < 256; off <<= 1) {
        int v = (t >= off) ? s[t - off] : 0;
        __syncthreads();
        s[t] += v;
        __syncthreads();
    }
    int run = (t == 0) ? 0 : s[t - 1];
    for (int i = beg; i < end; ++i) { offs[i] = run; run += sizes[i]; }
}

// ---------------- Phase 2c: per-graph weighted reduction (1 wave/graph) ----------------
__global__ __launch_bounds__(32) void ga_pass2(
    const float* __restrict__ g, const float* __restrict__ o,
    const int* __restrict__ offs, const int* __restrict__ sizes,
    const float* __restrict__ gmax, float* __restrict__ out, int n_graphs)
{
    int gr = blockIdx.x;
    if (gr >= n_graphs) return;
    int beg = offs[gr], cnt = sizes[gr];
    float mx = gmax[0];
    float se = 0.f, a0 = 0.f, a1 = 0.f;
    for (int i = threadIdx.x; i < cnt; i += 32) {
        long long idx = (long long)beg + i;
        float e = __expf(g[idx] - mx);
        v2f ov = *reinterpret_cast<const v2f*>(o + idx * 2);
        se += e; a0 += e * ov.x; a1 += e * ov.y;
    }
    #pragma unroll
    for (int off = 16; off > 0; off >>= 1) {
        se += __shfl_down(se, off, 32);
        a0 += __shfl_down(a0, off, 32);
        a1 += __shfl_down(a1, off, 32);
    }
    if (threadIdx.x == 0) {
        float inv = 1.0f / (se + 1e-16f);
        out[(long long)gr * 2 + 0] = a0 * inv;
        out[(long long)gr * 2 + 1] = a1 * inv;
    }
}

// ---------------------------------------------------------------------------
extern "C" void kernel_launch(void* const* d_in, const int* in_sizes, int n_in,
                              void* d_out, int out_size, void* d_ws, size_t ws_size,
                              hipStream_t stream)
{
    const float* states = (const float*)d_in[0];
    const int*   sizes  = (const int*)  d_in[1];
    const float* gate_w = (const float*)d_in[2];
    const float* gate_b = (const float*)d_in[3];
    const float* out_w  = (const float*)d_in[4];
    const float* out_b  = (const float*)d_in[5];
    float* out = (float*)d_out;

    const int n_nodes  = in_sizes[0] / D_FEAT;   // 500000
    const int n_graphs = in_sizes[1];            // 10000

    // Workspace layout (~6 MB total)
    char* ws = (char*)d_ws;
    size_t off = 0;
    float* gbuf = (float*)(ws + off); off += (size_t)n_nodes * 4;         // gate logits
    off = (off + 15) & ~(size_t)15;
    float* obuf = (float*)(ws + off); off += (size_t)n_nodes * 8;         // (o0,o1) pairs
    off = (off + 15) & ~(size_t)15;
    float* pmax = (float*)(ws + off); off += 1024 * 4;                    // block maxima
    float* gmax = (float*)(ws + off); off += 16;                          // global max
    int*   offs = (int*)  (ws + off); off += (size_t)n_graphs * 4;        // graph starts

    const int tiles   = (n_nodes + ROWS_PER_WAVE - 1) / ROWS_PER_WAVE;
    const int blocks1 = (tiles + WAVES_PER_BLOCK - 1) / WAVES_PER_BLOCK;

    ga_pass1<<<blocks1, 256, 0, stream>>>(states, gate_w, gate_b, out_w, out_b,
                                          gbuf, obuf, n_nodes);
    ga_max_partial<<<1024, 256, 0, stream>>>(gbuf, pmax, n_nodes);
    ga_max_final<<<1, 256, 0, stream>>>(pmax, gmax, 1024);
    ga_scan<<<1, 256, 0, stream>>>(sizes, offs, n_graphs);
    ga_pass2<<<n_graphs, 32, 0, stream>>>(gbuf, obuf, offs, sizes, gmax, out, n_graphs);
}